// Word2VecNS_27693949125158
// MI455X (gfx1250) — compile-verified
//
#include <hip/hip_runtime.h>

// Word2Vec negative-sampling scores: out[b][c] = dot(embed[targets[b]], embed[contexts[b][c]])
// B=16384, E=128, C=6. Bandwidth-bound gather (~57.7 MB @ 23.3 TB/s ~= 2.5us floor).
//
// Per wave32: one tile of 16 batch rows.
//  Phase 1: stage all 112 gathered rows (512B each) into LDS with
//           GLOBAL_LOAD_ASYNC_TO_LDS_B128 (1 instr = 1 coalesced row, ASYNCcnt,
//           no VGPRs) -> 112 row fetches in flight per wave.
//  Phase 2: K-loop reads 8B fragments from LDS (bank-conflict-free via 528B row
//           pitch) and accumulates via V_WMMA_F32_16X16X4_F32; keep the diagonal
//           of the 16x16 product (cross-batch terms are free — compute is ~10x
//           under the memory floor even with the 16x waste).

#define W2V_BATCH 16384
#define W2V_EMBED 128
#define W2V_NCTX  6
#define ROWS_PER_TILE (16 * (1 + W2V_NCTX)) /* 112 */
#define ROW_PITCH 132  /* dwords: 128 + 4 pad -> row stride mod 64 banks = 4 */

typedef float v2f __attribute__((ext_vector_type(2)));
typedef float v4f __attribute__((ext_vector_type(4)));
typedef float v8f __attribute__((ext_vector_type(8)));
typedef int   v4i __attribute__((ext_vector_type(4)));

// Async-to-LDS builtin takes (int4 AS1*, int4 AS3*, imm offset, imm cpol).
typedef __attribute__((address_space(1))) v4i* glb_v4i_p;
typedef __attribute__((address_space(3))) v4i* lds_v4i_p;

#if defined(__has_builtin)
#if __has_builtin(__builtin_amdgcn_global_load_async_to_lds_b128)
#define HAVE_ASYNC_LDS 1
#endif
#if __has_builtin(__builtin_amdgcn_s_wait_asynccnt)
#define HAVE_WAIT_ASYNC 1
#endif
#endif

__device__ __forceinline__ float vsel8(const v8f& v, int i) {
    switch (i) {
        case 0: return v[0]; case 1: return v[1];
        case 2: return v[2]; case 3: return v[3];
        case 4: return v[4]; case 5: return v[5];
        case 6: return v[6]; default: return v[7];
    }
}

__global__ __launch_bounds__(32) void
Word2VecNS_wmma_kernel(const int* __restrict__ targets,
                       const int* __restrict__ contexts,
                       const float* __restrict__ embed,
                       float* __restrict__ out) {
    __shared__ float lds[ROWS_PER_TILE * ROW_PITCH]; // 57.75 KB

    const int lane = threadIdx.x;      // 0..31 (wave32)
    const int m    = lane & 15;        // A-row (M) / B-col (N) index in tile
    const int koff = (lane >> 4) * 2;  // K sub-offset per ISA f32 16x4 layout
    const int b0   = blockIdx.x * 16;

    // ---------------- Phase 1: stage 112 rows into LDS ----------------
    // Row slots: [0..15] = target rows, [16 + t] = contexts[b0*6 + t] rows.
    // Indices are wave-uniform -> scalar loads; per-lane offset = lane*16B.
#if HAVE_ASYNC_LDS
#pragma unroll 4
    for (int r = 0; r < 16; ++r) {
        const float* src = embed + (long)targets[b0 + r] * W2V_EMBED + lane * 4;
        float* dst = &lds[r * ROW_PITCH + lane * 4];
        __builtin_amdgcn_global_load_async_to_lds_b128((glb_v4i_p)src, (lds_v4i_p)dst, 0, 0);
    }
#pragma unroll 4
    for (int t = 0; t < 16 * W2V_NCTX; ++t) {
        const float* src = embed + (long)contexts[b0 * W2V_NCTX + t] * W2V_EMBED + lane * 4;
        float* dst = &lds[(16 + t) * ROW_PITCH + lane * 4];
        __builtin_amdgcn_global_load_async_to_lds_b128((glb_v4i_p)src, (lds_v4i_p)dst, 0, 0);
    }
#if HAVE_WAIT_ASYNC
    __builtin_amdgcn_s_wait_asynccnt(0);
#else
    asm volatile("s_wait_asynccnt 0x0" ::: "memory");
#endif
    asm volatile("" ::: "memory"); // keep DS reads below the wait
#else
    // Fallback: batched sync staging, 8 coalesced row loads in flight.
    for (int r0 = 0; r0 < ROWS_PER_TILE; r0 += 8) {
        v4f tmp[8];
#pragma unroll
        for (int i = 0; i < 8; ++i) {
            const int r = r0 + i;
            const long idx = (r < 16) ? (long)targets[b0 + r]
                                      : (long)contexts[b0 * W2V_NCTX + (r - 16)];
            tmp[i] = *(const v4f*)(embed + idx * W2V_EMBED + lane * 4);
        }
#pragma unroll
        for (int i = 0; i < 8; ++i)
            *(v4f*)&lds[(r0 + i) * ROW_PITCH + lane * 4] = tmp[i];
    }
#endif

    // ---------------- Phase 2: WMMA K-loop from LDS ----------------
    // Per lane: A frag = {A[m][k+koff], A[m][k+koff+1]} from row slot m,
    //           B_j frag = ctx row slot 16 + n*6 + j at same K offsets.
    const float* arow = &lds[m * ROW_PITCH + koff];
    const float* brow[W2V_NCTX];
#pragma unroll
    for (int j = 0; j < W2V_NCTX; ++j)
        brow[j] = &lds[(16 + m * W2V_NCTX + j) * ROW_PITCH + koff];

    v8f acc[W2V_NCTX] = {};
#pragma unroll 4
    for (int k = 0; k < W2V_EMBED; k += 4) {
        v2f a = *(const v2f*)(arow + k);
#pragma unroll
        for (int j = 0; j < W2V_NCTX; ++j) {
            v2f b = *(const v2f*)(brow[j] + k);
            // 8 args: (neg_a, A, neg_b, B, c_mod, C, reuse_a, reuse_b)
            acc[j] = __builtin_amdgcn_wmma_f32_16x16x4_f32(
                false, a, false, b, (short)0, acc[j], false, false);
        }
    }

    // Diagonal of 16x16 f32 D: (d,d) lives in lanes 0..7 (d=lane, vgpr=lane)
    // and lanes 24..31 (d=lane-16, vgpr=lane-24).
    const bool active  = (lane < 8) || (lane >= 24);
    const int  diagRow = (lane < 8) ? lane : (lane - 16);
    const int  vidx    = lane & 7;
    if (active) {
#pragma unroll
        for (int j = 0; j < W2V_NCTX; ++j)
            out[(b0 + diagRow) * W2V_NCTX + j] = vsel8(acc[j], vidx);
    }
}

extern "C" void kernel_launch(void* const* d_in, const int* in_sizes, int n_in,
                              void* d_out, int out_size, void* d_ws, size_t ws_size,
                              hipStream_t stream) {
    (void)in_sizes; (void)n_in; (void)out_size; (void)d_ws; (void)ws_size;
    const int*   targets  = (const int*)d_in[0];   // [16384]
    const int*   contexts = (const int*)d_in[1];   // [16384, 6]
    const float* embed    = (const float*)d_in[2]; // [200000, 128]
    float*       out      = (float*)d_out;         // [16384, 6]

    dim3 grid(W2V_BATCH / 16);
    dim3 block(32);
    Word2VecNS_wmma_kernel<<<grid, block, 0, stream>>>(targets, contexts, embed, out);
}